// SigmaElasticity_4166118277758
// MI455X (gfx1250) — compile-verified
//
#include <hip/hip_runtime.h>

#define TILE 256
#define SWEEPS 3
#define USE_ASYNC_LDS 1

// Hardware-approximate (≈1 ulp) reciprocal / rsqrt / sqrt / log2 — exactly
// what Jacobi rotations need; avoids hipcc's ~10-instruction IEEE div fixups.
__device__ __forceinline__ float h_rcp(float x)  { return __builtin_amdgcn_rcpf(x); }
__device__ __forceinline__ float h_rsq(float x)  { return __builtin_amdgcn_rsqf(x); }
__device__ __forceinline__ float h_sqrt(float x) { return __builtin_amdgcn_sqrtf(x); }
__device__ __forceinline__ float h_log2(float x) { return __builtin_amdgcn_logf(x); }

// Branchless Givens rotation zeroing a_pq of symmetric [[app,apq],[apq,aqq]].
// Returns c, s and t = s/c (diagonal update: app' = app - t*apq, aqq' += t*apq).
__device__ __forceinline__ void givens(float app, float aqq, float apq,
                                       float& c, float& s, float& t) {
    float az   = __builtin_fabsf(apq);
    // apq==0 -> rcp=+inf -> zeta=+-inf (or NaN if diff==0); both selected away.
    float zeta = (aqq - app) * (0.5f * h_rcp(apq));
    float tt   = __builtin_copysignf(1.0f, zeta) *
                 h_rcp(__builtin_fabsf(zeta) +
                       h_sqrt(__builtin_fmaf(zeta, zeta, 1.0f)));
    tt = (az > 0.0f) ? tt : 0.0f;
    float cc = h_rsq(__builtin_fmaf(tt, tt, 1.0f));
    c = cc;
    s = tt * cc;
    t = tt;
}

__global__ __launch_bounds__(TILE)
void sigma_elasticity_kernel(const float* __restrict__ F,
                             const float* __restrict__ logE_p,
                             const float* __restrict__ nu_p,
                             float* __restrict__ out,
                             int n) {
    __shared__ float tile[TILE * 9];
    const int t      = threadIdx.x;
    const int nTiles = (n + TILE - 1) / TILE;
    const int total  = n * 9;

    // Uniform scalars: folded to SALU-float ops by the compiler.
    const float E  = expf(logE_p[0]);
    const float nu = nu_p[0];
    const float mu = E / (2.0f * (1.0f + nu));
    const float la = E * nu / ((1.0f + nu) * (1.0f - 2.0f * nu));

    for (int tb = blockIdx.x; tb < nTiles; tb += gridDim.x) {
        const int base = tb * (TILE * 9);

        // Prefetch next tile (global_prefetch_b8): 256 lanes * 36B stride
        // covers the whole 9216B region.
        const int nb = tb + (int)gridDim.x;
        if (nb < nTiles)
            __builtin_prefetch(F + (size_t)nb * (TILE * 9) + (size_t)t * 9, 0, 3);

        const bool full = (base + TILE * 9) <= total;
#if USE_ASYNC_LDS
        if (full) {
            // Async global -> LDS copy (ASYNCcnt), 128B/wave per instruction.
            const float* gsrc = F + base;
#pragma unroll
            for (int k = 0; k < 9; ++k) {
                const int idx = t + k * TILE;
                unsigned lds_off = (unsigned)(unsigned long long)&tile[idx];
                asm volatile("global_load_async_to_lds_b32 %0, %1, off"
                             :: "v"(lds_off), "v"(gsrc + idx) : "memory");
            }
            asm volatile("s_wait_asynccnt 0" ::: "memory");
        } else
#endif
        {
#pragma unroll
            for (int k = 0; k < 9; ++k) {
                const int idx = t + k * TILE;
                const int g   = base + idx;
                tile[idx] = (g < total) ? F[g] : 1.0f;  // finite filler for tail
            }
        }
        __syncthreads();

        // Per-thread 3x3: conflict-free LDS reads (stride 9, gcd(9,64)=1).
        const int me = t * 9;
        const float f0 = tile[me + 0], f1 = tile[me + 1], f2 = tile[me + 2];
        const float f3 = tile[me + 3], f4 = tile[me + 4], f5 = tile[me + 5];
        const float f6 = tile[me + 6], f7 = tile[me + 7], f8 = tile[me + 8];

        // A = F * F^T (symmetric); eig(A) = (U, sigma^2) of the SVD of F.
        float a00 = f0 * f0 + f1 * f1 + f2 * f2;
        float a01 = f0 * f3 + f1 * f4 + f2 * f5;
        float a02 = f0 * f6 + f1 * f7 + f2 * f8;
        float a11 = f3 * f3 + f4 * f4 + f5 * f5;
        float a12 = f3 * f6 + f4 * f7 + f5 * f8;
        float a22 = f6 * f6 + f7 * f7 + f8 * f8;

        float v00 = 1.f, v01 = 0.f, v02 = 0.f;
        float v10 = 0.f, v11 = 1.f, v12 = 0.f;
        float v20 = 0.f, v21 = 0.f, v22 = 1.f;

#pragma unroll
        for (int sw = 0; sw < SWEEPS; ++sw) {
            float c, s, tt, x, r0, r1;
            // pair (0,1)
            givens(a00, a11, a01, c, s, tt);
            a00 -= tt * a01; a11 += tt * a01; a01 = 0.0f;
            r0 = a02; r1 = a12;
            a02 = c * r0 - s * r1;  a12 = s * r0 + c * r1;
            x = v00; v00 = c * x - s * v01; v01 = s * x + c * v01;
            x = v10; v10 = c * x - s * v11; v11 = s * x + c * v11;
            x = v20; v20 = c * x - s * v21; v21 = s * x + c * v21;
            // pair (0,2)
            givens(a00, a22, a02, c, s, tt);
            a00 -= tt * a02; a22 += tt * a02; a02 = 0.0f;
            r0 = a01; r1 = a12;
            a01 = c * r0 - s * r1;  a12 = s * r0 + c * r1;
            x = v00; v00 = c * x - s * v02; v02 = s * x + c * v02;
            x = v10; v10 = c * x - s * v12; v12 = s * x + c * v12;
            x = v20; v20 = c * x - s * v22; v22 = s * x + c * v22;
            // pair (1,2)
            givens(a11, a22, a12, c, s, tt);
            a11 -= tt * a12; a22 += tt * a12; a12 = 0.0f;
            r0 = a01; r1 = a02;
            a01 = c * r0 - s * r1;  a02 = s * r0 + c * r1;
            x = v01; v01 = c * x - s * v02; v02 = s * x + c * v02;
            x = v11; v11 = c * x - s * v12; v12 = s * x + c * v12;
            x = v21; v21 = c * x - s * v22; v22 = s * x + c * v22;
        }

        // eps = log(max(sigma,1e-5)) = 0.5*ln(max(lambda,1e-10))
        //     = log2(max(lambda,1e-10)) * (0.5*ln2).  Raw v_log_f32 is safe:
        // argument is clamped >= 1e-10 (normal range).
        const float lmin   = 1e-10f;
        const float hln2   = 0.34657359027997264f;  // 0.5 * ln(2)
        const float e0 = hln2 * h_log2(fmaxf(a00, lmin));
        const float e1 = hln2 * h_log2(fmaxf(a11, lmin));
        const float e2 = hln2 * h_log2(fmaxf(a22, lmin));
        const float tr = e0 + e1 + e2;
        const float t0 = 2.0f * mu * e0 + la * tr;
        const float t1 = 2.0f * mu * e1 + la * tr;
        const float t2 = 2.0f * mu * e2 + la * tr;

        // stress = sum_j tau_j * u_j u_j^T  (symmetric, 6 unique entries)
        const float s00 = t0 * v00 * v00 + t1 * v01 * v01 + t2 * v02 * v02;
        const float s01 = t0 * v00 * v10 + t1 * v01 * v11 + t2 * v02 * v12;
        const float s02 = t0 * v00 * v20 + t1 * v01 * v21 + t2 * v02 * v22;
        const float s11 = t0 * v10 * v10 + t1 * v11 * v11 + t2 * v12 * v12;
        const float s12 = t0 * v10 * v20 + t1 * v11 * v21 + t2 * v12 * v22;
        const float s22 = t0 * v20 * v20 + t1 * v21 * v21 + t2 * v22 * v22;

        tile[me + 0] = s00; tile[me + 1] = s01; tile[me + 2] = s02;
        tile[me + 3] = s01; tile[me + 4] = s11; tile[me + 5] = s12;
        tile[me + 6] = s02; tile[me + 7] = s12; tile[me + 8] = s22;
        __syncthreads();

        // Coalesced store through LDS.
#pragma unroll
        for (int k = 0; k < 9; ++k) {
            const int idx = t + k * TILE;
            const int g   = base + idx;
            if (g < total) out[g] = tile[idx];
        }
        __syncthreads();  // protect tile before next iteration's loads
    }
}

extern "C" void kernel_launch(void* const* d_in, const int* in_sizes, int n_in,
                              void* d_out, int out_size, void* d_ws, size_t ws_size,
                              hipStream_t stream) {
    (void)n_in; (void)out_size; (void)d_ws; (void)ws_size;
    const float* F    = (const float*)d_in[0];
    const float* logE = (const float*)d_in[1];
    const float* nu   = (const float*)d_in[2];
    float*       out  = (float*)d_out;

    const int n      = in_sizes[0] / 9;
    const int nTiles = (n + TILE - 1) / TILE;
    int blocks = nTiles < 4096 ? nTiles : 4096;
    if (blocks < 1) blocks = 1;

    sigma_elasticity_kernel<<<dim3(blocks), dim3(TILE), 0, stream>>>(
        F, logE, nu, out, n);
}